// RNNModel_26757646254131
// MI455X (gfx1250) — compile-verified
//
#include <hip/hip_runtime.h>
#include <math.h>

// CDNA5 / gfx1250, wave32.
typedef __attribute__((ext_vector_type(2))) float v2f;
typedef __attribute__((ext_vector_type(8))) float v8f;

// Branch-free tanh: tanh(x) = 1 - 2/(exp(2x)+1)
//   exp via v_exp_f32 (base-2, pre-scaled), reciprocal via v_rcp_f32.
//   Saturates exactly: x->+inf => exp=inf => rcp=0 => 1; x->-inf => exp=0 => -1.
__device__ __forceinline__ float tanh_fast(float x) {
    const float LOG2E_X2 = 2.8853900817779268f;   // 2*log2(e)
    float e = __builtin_amdgcn_exp2f(x * LOG2E_X2);
    float r = __builtin_amdgcn_rcpf(e + 1.0f);
    return __builtin_fmaf(-2.0f, r, 1.0f);
}

// Fused RNN: per block (1 wave, 32 lanes) handle 16 locations over all T steps.
// Recurrence GEMM [16x4]@[4x4] done as transposed WMMA: D = Wh^T (16x4 A) x c^T (4x16 B) + pre^T (C).
//   A layout (f32 16x4): lanes 0-15 vgpr{0,1} = K{0,1}; lanes 16-31 = K{2,3}  (per ISA doc)
//   B layout (f32 4x16): mirrored half-wave K split: vgpr0 = rows {0|2}, vgpr1 = rows {1|3}
//   C/D layout: vgpr j: lanes 0-15 -> row j, lanes 16-31 -> row j+8; col = lane&15 (location)
__global__ __launch_bounds__(32)
void rnn_fused_scan(const float* __restrict__ x,     // [L,T,8]
                    const float* __restrict__ emb,   // [L,4]
                    const float* __restrict__ W1g,   // [12,4]
                    const float* __restrict__ b1g,   // [4]
                    const float* __restrict__ Wpg,   // [4,4]
                    const float* __restrict__ bpg,   // [4]
                    const float* __restrict__ Wig,   // [4,4]
                    const float* __restrict__ big,   // [4]
                    const float* __restrict__ Whg,   // [4,4]
                    const float* __restrict__ Wo1g,  // [4,6]
                    const float* __restrict__ bo1g,  // [6]
                    const float* __restrict__ Wo2g,  // [6,1]
                    const float* __restrict__ bo2g,  // [1]
                    float* __restrict__ out,         // [L,T]
                    int T)
{
    const int lane = threadIdx.x;          // 0..31
    const int m    = lane & 15;            // location within tile / matrix column
    const int loc  = blockIdx.x * 16 + m;  // lanes 16-31 mirror lanes 0-15 (safe dup loads)
    const bool lo  = (lane < 16);

    // ---- small weights into registers (uniform -> scalar loads) ----
    float w1[12][4];
#pragma unroll
    for (int r = 0; r < 12; ++r)
#pragma unroll
        for (int c = 0; c < 4; ++c) w1[r][c] = W1g[r * 4 + c];

    float bb1[4];
#pragma unroll
    for (int c = 0; c < 4; ++c) bb1[c] = b1g[c];

    // Fold Wp@Wi (no nonlinearity between them): w2 = Wp*Wi, b2 = bp*Wi + bi
    float w2[4][4], b2[4];
#pragma unroll
    for (int c = 0; c < 4; ++c) {
        float acc = big[c];
#pragma unroll
        for (int k = 0; k < 4; ++k) acc += bpg[k] * Wig[k * 4 + c];
        b2[c] = acc;
    }
#pragma unroll
    for (int r = 0; r < 4; ++r)
#pragma unroll
        for (int c = 0; c < 4; ++c) {
            float acc = 0.0f;
#pragma unroll
            for (int k = 0; k < 4; ++k) acc += Wpg[r * 4 + k] * Wig[k * 4 + c];
            w2[r][c] = acc;
        }

    float wo1[4][6];
#pragma unroll
    for (int h = 0; h < 4; ++h)
#pragma unroll
        for (int j = 0; j < 6; ++j) wo1[h][j] = Wo1g[h * 6 + j];
    float vbo1[6], vwo2[6];
#pragma unroll
    for (int j = 0; j < 6; ++j) { vbo1[j] = bo1g[j]; vwo2[j] = Wo2g[j]; }
    const float sbo2 = bo2g[0];

    // ---- constant WMMA A matrix = Wh^T (16x4, rows >=4 zero) ----
    const int k0 = lo ? 0 : 2;
    v2f amat;
    amat.x = (m < 4) ? Whg[(k0 + 0) * 4 + m] : 0.0f;  // A[m, k0]   = Wh[k0,   m]
    amat.y = (m < 4) ? Whg[(k0 + 1) * 4 + m] : 0.0f;  // A[m, k0+1] = Wh[k0+1, m]

    v2f bmat;                  // carry^T, zero init
    bmat.x = 0.0f; bmat.y = 0.0f;

    // location embedding (per-lane constant over time)
    const float e0 = emb[loc * 4 + 0];
    const float e1 = emb[loc * 4 + 1];
    const float e2 = emb[loc * 4 + 2];
    const float e3 = emb[loc * 4 + 3];

    const float* xp = x + (size_t)loc * (size_t)T * 8;
    float* yp = out + (size_t)loc * (size_t)T;

    for (int t0 = 0; t0 < T; t0 += 4) {
        // keep the per-lane x stream ~1KB ahead in cache (global_prefetch_b8)
        __builtin_prefetch(xp + (size_t)t0 * 8 + 256, 0, 1);

        // ---- phase 1: chain-independent pre-projection for the next 4 steps ----
        float pre[4][4];
#pragma unroll
        for (int u = 0; u < 4; ++u) {
            const float4 xa = *(const float4*)(xp + (size_t)(t0 + u) * 8);
            const float4 xb = *(const float4*)(xp + (size_t)(t0 + u) * 8 + 4);
            const float in12[12] = {xa.x, xa.y, xa.z, xa.w,
                                    xb.x, xb.y, xb.z, xb.w,
                                    e0, e1, e2, e3};
            float h4[4];
#pragma unroll
            for (int c = 0; c < 4; ++c) {
                float acc = bb1[c];
#pragma unroll
                for (int r = 0; r < 12; ++r) acc += in12[r] * w1[r][c];
                h4[c] = fmaxf(acc, 0.0f);              // relu
            }
#pragma unroll
            for (int c = 0; c < 4; ++c) {
                float acc = b2[c];
#pragma unroll
                for (int k = 0; k < 4; ++k) acc += h4[k] * w2[k][c];
                pre[u][c] = lo ? acc : 0.0f;           // zero rows 8-11 of C
            }
        }

        // ---- phase 2: 4 chained recurrence steps (WMMA + tanh + head) ----
        float yarr[4];
#pragma unroll
        for (int u = 0; u < 4; ++u) {
            v8f cm;
            cm[0] = pre[u][0]; cm[1] = pre[u][1];
            cm[2] = pre[u][2]; cm[3] = pre[u][3];
            cm[4] = 0.0f; cm[5] = 0.0f; cm[6] = 0.0f; cm[7] = 0.0f;

            // D = Wh^T x c^T + pre^T    (v_wmma_f32_16x16x4_f32)
            v8f d = __builtin_amdgcn_wmma_f32_16x16x4_f32(
                false, amat, false, bmat, (short)0, cm, false, false);

            const float c0 = tanh_fast(d[0]);
            const float c1 = tanh_fast(d[1]);
            const float c2 = tanh_fast(d[2]);
            const float c3 = tanh_fast(d[3]);

            // repack rows {2,3} into upper half-wave for next-step B
            const float s2 = __shfl(c2, m, 32);
            const float s3 = __shfl(c3, m, 32);
            bmat.x = lo ? c0 : s2;
            bmat.y = lo ? c1 : s3;

            // output head: relu(c@Wo1 + bo1) @ Wo2 + bo2  (valid on lanes 0-15)
            float y = sbo2;
#pragma unroll
            for (int j = 0; j < 6; ++j) {
                float o = vbo1[j];
                o += c0 * wo1[0][j];
                o += c1 * wo1[1][j];
                o += c2 * wo1[2][j];
                o += c3 * wo1[3][j];
                o = fmaxf(o, 0.0f);
                y += o * vwo2[j];
            }
            yarr[u] = y;
        }

        if (lo) {
            float4 yv;
            yv.x = yarr[0]; yv.y = yarr[1]; yv.z = yarr[2]; yv.w = yarr[3];
            *(float4*)(yp + t0) = yv;
        }
    }
}

extern "C" void kernel_launch(void* const* d_in, const int* in_sizes, int n_in,
                              void* d_out, int out_size, void* d_ws, size_t ws_size,
                              hipStream_t stream) {
    const float* x   = (const float*)d_in[0];
    const float* emb = (const float*)d_in[1];
    const float* W1  = (const float*)d_in[2];
    const float* b1  = (const float*)d_in[3];
    const float* Wp  = (const float*)d_in[4];
    const float* bp  = (const float*)d_in[5];
    const float* Wi  = (const float*)d_in[6];
    const float* bi  = (const float*)d_in[7];
    const float* Wh  = (const float*)d_in[8];
    const float* Wo1 = (const float*)d_in[9];
    const float* bo1 = (const float*)d_in[10];
    const float* Wo2 = (const float*)d_in[11];
    const float* bo2 = (const float*)d_in[12];
    float* out = (float*)d_out;

    const int L = in_sizes[1] / 4;            // emb is [L,4]
    const int T = in_sizes[0] / (L * 8);      // x is [L,T,8]

    dim3 grid(L / 16), block(32);
    rnn_fused_scan<<<grid, block, 0, stream>>>(x, emb, W1, b1, Wp, bp, Wi, bi,
                                               Wh, Wo1, bo1, Wo2, bo2, out, T);
}